// SimpleLSTM_45810121179433
// MI455X (gfx1250) — compile-verified
//
#include <hip/hip_runtime.h>

// LSTM, B=32 T=512 I=H=1024, fp32. CDNA5 (gfx1250) wave32 + V_WMMA_F32_16X16X4_F32.
//
//  - x_gates = x @ W_ih^T + b_ih + b_hh precomputed as one big parallel GEMM
//    (if ws_size fits 268MB; otherwise fused into each step).
//  - One fused kernel per timestep: block = 8 waves, each wave computes ONE
//    16x16 (M-tile, gate) product over K via fp32 WMMA (256 WMMAs/wave ->
//    short critical path for the sequential scan), gates exchanged through
//    LDS, cell update elementwise, h/c written straight to d_out.
//  - Running c state lives in the c_n slot of d_out; h_{t-1} is read from the
//    output tensor -> zero workspace needed for the scan itself.

typedef __attribute__((ext_vector_type(2))) float v2f;
typedef __attribute__((ext_vector_type(8))) float v8f;

#define BB 32
#define TT 512
#define II 1024
#define HH 1024
#define G4 4096

static __device__ __forceinline__ v8f wmma4(v2f a, v2f b, v8f c) {
  // D = A(16x4 f32) * B(4x16 f32) + C(16x16 f32)
  return __builtin_amdgcn_wmma_f32_16x16x4_f32(
      /*neg_a=*/false, a, /*neg_b=*/false, b,
      /*c_mod=*/(short)0, c, /*reuse_a=*/false, /*reuse_b=*/false);
}

static __device__ __forceinline__ v2f ld2(const float* p) {
  return *(const v2f*)p;
}

static __device__ __forceinline__ float sigmoid_f(float x) {
  return 1.0f / (1.0f + __expf(-x));
}

static __device__ __forceinline__ float tanh_f(float x) {
  x = fminf(fmaxf(x, -15.0f), 15.0f);
  float e = __expf(2.0f * x);
  return (e - 1.0f) / (e + 1.0f);
}

// ---------------------------------------------------------------------------
// x_gates GEMM: C[m][n] = sum_k x[m][k]*W_ih[n][k] + b_ih[n] + b_hh[n]
// M = B*T = 16384, N = 4H = 4096, K = I = 1024.
// Wave tile: 64x16 (4 M-subtiles). Block = 8 waves -> 64x128 per block.
// Grid: (N/128, M/64) = (32, 256).
// ---------------------------------------------------------------------------
__global__ __launch_bounds__(256)
void xg_gemm_kernel(const float* __restrict__ x, const float* __restrict__ W_ih,
                    const float* __restrict__ b_ih, const float* __restrict__ b_hh,
                    float* __restrict__ xg) {
  const int lane = threadIdx.x & 31;
  const int wid  = threadIdx.x >> 5;
  const int nrow = lane & 15;           // M index (A) / N index (B, C)
  const int koff = (lane >> 4) << 1;    // lanes 16-31 carry K = k0+2, k0+3
  const int mhi  = (lane >> 4) << 3;    // C layout: high lanes -> M = r + 8

  const int col = blockIdx.x * 128 + wid * 16 + nrow;   // 0..4095
  const int m0  = blockIdx.y * 64;

  v8f acc[4];
#pragma unroll
  for (int i = 0; i < 4; ++i) acc[i] = {};

  const float* wrow  = W_ih + (size_t)col * II;
  const float* arow0 = x + (size_t)(m0 + nrow) * II;

  for (int k0 = 0; k0 < II; k0 += 4) {
    v2f b = ld2(wrow + k0 + koff);
#pragma unroll
    for (int mt = 0; mt < 4; ++mt) {
      v2f a = ld2(arow0 + (size_t)mt * 16 * II + k0 + koff);
      acc[mt] = wmma4(a, b, acc[mt]);
    }
  }

  const float bias = b_ih[col] + b_hh[col];
#pragma unroll
  for (int mt = 0; mt < 4; ++mt) {
#pragma unroll
    for (int r = 0; r < 8; ++r) {
      int m = m0 + mt * 16 + mhi + r;
      xg[(size_t)m * G4 + col] = acc[mt][r] + bias;
    }
  }
}

// ---------------------------------------------------------------------------
// One LSTM step.
// Block = 256 threads (8 waves), handles 16 columns of H for all 4 gates and
// all 32 batch rows. Wave w: gate = w&3, M-tile = w>>2 -> one 16x16 WMMA
// accumulator per wave (256 chained WMMAs over K=1024). Gate tiles exchanged
// through LDS, then 256 threads do the cell update (2 elements each).
// Grid: H/16 = 64 blocks -> 512 waves per step.
// ---------------------------------------------------------------------------
__global__ __launch_bounds__(256)
void lstm_step_kernel(const float* __restrict__ xg,   // null => fuse x-projection
                      const float* __restrict__ x,
                      const float* __restrict__ h_prev, int ldh,
                      const float* __restrict__ c_prev,
                      const float* __restrict__ W_ih,
                      const float* __restrict__ W_hh,
                      const float* __restrict__ b_ih,
                      const float* __restrict__ b_hh,
                      float* __restrict__ out,
                      float* __restrict__ c_out,
                      float* __restrict__ h_final,    // null except last step
                      int t) {
  __shared__ float gtile[8 * 256];   // 8 waves x (16x16 f32 tile), 8 KB

  const int tid  = threadIdx.x;
  const int lane = tid & 31;
  const int wid  = tid >> 5;
  const int gate = wid & 3;          // i, f, g, o
  const int mt   = wid >> 2;         // batch tile 0 / 1
  const int nrow = lane & 15;
  const int koff = (lane >> 4) << 1; // lanes 16-31 carry K = k0+2, k0+3

  const int n0  = blockIdx.x * 16;   // column group within H
  const int col = n0 + nrow;

  v8f acc = {};

  // gates[gate] tile += h_prev[mt*16.., :] @ W_hh[gate*H + n0.., :]^T
  {
    const float* arow = h_prev + (size_t)(mt * 16 + nrow) * ldh;
    const float* brow = W_hh + (size_t)(gate * HH + col) * HH;
    for (int k0 = 0; k0 < HH; k0 += 8) {
      v2f a0 = ld2(arow + k0 + koff);
      v2f b0 = ld2(brow + k0 + koff);
      v2f a1 = ld2(arow + k0 + 4 + koff);
      v2f b1 = ld2(brow + k0 + 4 + koff);
      acc = wmma4(a0, b0, acc);
      acc = wmma4(a1, b1, acc);
    }
  }

  // Fused x-projection when x_gates was not precomputed
  if (!xg) {
    const float* arow = x + ((size_t)(mt * 16 + nrow) * TT + t) * II;
    const float* brow = W_ih + (size_t)(gate * HH + col) * II;
    for (int k0 = 0; k0 < II; k0 += 8) {
      v2f a0 = ld2(arow + k0 + koff);
      v2f b0 = ld2(brow + k0 + koff);
      v2f a1 = ld2(arow + k0 + 4 + koff);
      v2f b1 = ld2(brow + k0 + 4 + koff);
      acc = wmma4(a0, b0, acc);
      acc = wmma4(a1, b1, acc);
    }
  }

  // Dump the 16x16 tile to LDS in register order:
  // element (M = r + 8*(lane>=16), N = lane&15) at [wid*256 + r*32 + lane].
#pragma unroll
  for (int r = 0; r < 8; ++r) gtile[wid * 256 + r * 32 + lane] = acc[r];
  __syncthreads();

  // Cell update: 32 batches x 16 cols = 512 elements, 2 per thread.
#pragma unroll
  for (int e0 = 0; e0 < 2; ++e0) {
    const int e    = e0 * 256 + tid;
    const int b    = e >> 4;                 // batch 0..31
    const int c    = e & 15;                 // column within group
    const int emt  = b >> 4;                 // which M-tile
    const int mloc = b & 15;                 // row within tile
    const int li   = (mloc & 7) * 32 + c + ((mloc & 8) ? 16 : 0);

    float gi = gtile[(emt * 4 + 0) * 256 + li];
    float gf = gtile[(emt * 4 + 1) * 256 + li];
    float gg = gtile[(emt * 4 + 2) * 256 + li];
    float go = gtile[(emt * 4 + 3) * 256 + li];

    const int ocol = n0 + c;
    if (xg) {
      const float* p = xg + ((size_t)b * TT + t) * G4 + ocol;
      gi += p[0]; gf += p[HH]; gg += p[2 * HH]; go += p[3 * HH];
    } else {
      gi += b_ih[ocol] + b_hh[ocol];
      gf += b_ih[HH + ocol] + b_hh[HH + ocol];
      gg += b_ih[2 * HH + ocol] + b_hh[2 * HH + ocol];
      go += b_ih[3 * HH + ocol] + b_hh[3 * HH + ocol];
    }

    const float iv = sigmoid_f(gi);
    const float fv = sigmoid_f(gf);
    const float gv = tanh_f(gg);
    const float ov = sigmoid_f(go);
    const float cold = c_prev[(size_t)b * HH + ocol];
    const float cnew = fv * cold + iv * gv;
    const float hnew = ov * tanh_f(cnew);

    out[((size_t)b * TT + t) * HH + ocol] = hnew;
    c_out[(size_t)b * HH + ocol] = cnew;
    if (h_final) h_final[(size_t)b * HH + ocol] = hnew;
  }
}

// ---------------------------------------------------------------------------
extern "C" void kernel_launch(void* const* d_in, const int* in_sizes, int n_in,
                              void* d_out, int out_size, void* d_ws, size_t ws_size,
                              hipStream_t stream) {
  const float* x    = (const float*)d_in[0];
  const float* h0   = (const float*)d_in[1];
  const float* c0   = (const float*)d_in[2];
  const float* W_ih = (const float*)d_in[3];
  const float* W_hh = (const float*)d_in[4];
  const float* b_ih = (const float*)d_in[5];
  const float* b_hh = (const float*)d_in[6];

  float* out   = (float*)d_out;                        // (B, T, H)
  float* h_n   = out + (size_t)BB * TT * HH;           // (B, H)
  float* c_reg = h_n + (size_t)BB * HH;                // (B, H): running c -> c_n

  const size_t xg_bytes = (size_t)BB * TT * G4 * sizeof(float);
  float* xg = (ws_size >= xg_bytes) ? (float*)d_ws : nullptr;

  if (xg) {
    xg_gemm_kernel<<<dim3(G4 / 128, (BB * TT) / 64), 256, 0, stream>>>(
        x, W_ih, b_ih, b_hh, xg);
  }

  for (int t = 0; t < TT; ++t) {
    const float* hp = (t == 0) ? h0 : (out + (size_t)(t - 1) * HH);
    const int    ldh = (t == 0) ? HH : TT * HH;
    const float* cp = (t == 0) ? c0 : c_reg;
    lstm_step_kernel<<<dim3(HH / 16), 256, 0, stream>>>(
        xg, x, hp, ldh, cp, W_ih, W_hh, b_ih, b_hh,
        out, c_reg, (t == TT - 1) ? h_n : nullptr, t);
  }
}